// SO3net_76957224010213
// MI455X (gfx1250) — compile-verified
//
#include <hip/hip_runtime.h>
#include <math.h>

#define N_ATOMS 5000
#define N_EDGES 100000
#define NF      64
#define SDIM    9
#define NRBF    20
#define CUT_R   5.0f
#define ROWSZ   (SDIM * NF)   /* 576 floats per atom */

typedef __attribute__((ext_vector_type(2))) float v2f;
typedef __attribute__((ext_vector_type(8))) float v8f;

__device__ __forceinline__ v8f wmma4(v2f a, v2f b, v8f c) {
  // V_WMMA_F32_16X16X4_F32 : D(16x16 f32) = A(16x4 f32) * B(4x16 f32) + C
  return __builtin_amdgcn_wmma_f32_16x16x4_f32(false, a, false, b, (short)0, c, false, false);
}

// Real-basis Clebsch-Gordan contraction for lmax=2 (83 nonzeros, fully unrolled).
// o[c] = sum_{a,b} CG[a,b,c] * A[a] * B[b]
__device__ __forceinline__ void cg_contract(const float* a, const float* b, float* o) {
  const float s3 = 0.5773502691896258f;   // 1/sqrt(3)
  const float s5 = 0.4472135954999579f;   // 1/sqrt(5)
  const float A0 = 0.8164965809277260f;   // sqrt(2/3)
  const float A1 = 0.4082482904638630f;   // 1/sqrt(6)
  const float A2 = 0.7071067811865476f;   // 1/sqrt(2)
  const float B0 = 0.6324555320336759f;   // 2/sqrt(10)
  const float B1 = 0.3162277660168379f;   // 1/sqrt(10)
  const float B2 = 0.5477225575051661f;   // sqrt(3/10)
  const float D0 = 0.5345224838248488f;   // sqrt(2/7)
  const float D1 = 0.2672612419124244f;   // 1/sqrt(14)
  const float D2 = 0.4629100498862757f;   // sqrt(3/14)
  o[0] = a[0]*b[0] - s3*(a[1]*b[1] + a[2]*b[2] + a[3]*b[3])
       + s5*(a[4]*b[4] + a[5]*b[5] + a[6]*b[6] + a[7]*b[7] + a[8]*b[8]);
  o[1] = a[0]*b[1] + a[1]*b[0] + B1*(a[6]*b[1] + a[1]*b[6]) + B2*(a[8]*b[1] + a[1]*b[8])
       - B2*(a[5]*b[2] + a[2]*b[5]) - B2*(a[4]*b[3] + a[3]*b[4]);
  o[2] = a[0]*b[2] + a[2]*b[0] - B0*(a[6]*b[2] + a[2]*b[6])
       - B2*(a[7]*b[3] + a[3]*b[7]) - B2*(a[5]*b[1] + a[1]*b[5]);
  o[3] = a[0]*b[3] + a[3]*b[0] + B1*(a[6]*b[3] + a[3]*b[6]) - B2*(a[7]*b[2] + a[2]*b[7])
       - B2*(a[8]*b[3] + a[3]*b[8]) - B2*(a[4]*b[1] + a[1]*b[4]);
  o[4] = a[0]*b[4] + a[4]*b[0] + A2*(a[1]*b[3] + a[3]*b[1])
       + D0*(a[6]*b[4] + a[4]*b[6]) - D2*(a[7]*b[5] + a[5]*b[7]);
  o[5] = a[0]*b[5] + a[5]*b[0] + A2*(a[1]*b[2] + a[2]*b[1]) - D1*(a[6]*b[5] + a[5]*b[6])
       + D2*(a[8]*b[5] + a[5]*b[8]) - D2*(a[7]*b[4] + a[4]*b[7]);
  o[6] = a[0]*b[6] + a[6]*b[0] + A0*a[2]*b[2] - A1*(a[1]*b[1] + a[3]*b[3])
       - D0*a[6]*b[6] - D1*(a[7]*b[7] + a[5]*b[5]) + D0*(a[8]*b[8] + a[4]*b[4]);
  o[7] = a[0]*b[7] + a[7]*b[0] + A2*(a[2]*b[3] + a[3]*b[2]) - D1*(a[6]*b[7] + a[7]*b[6])
       - D2*(a[7]*b[8] + a[8]*b[7]) - D2*(a[5]*b[4] + a[4]*b[5]);
  o[8] = a[0]*b[8] + a[8]*b[0] + A2*(a[3]*b[3] - a[1]*b[1])
       + D0*(a[6]*b[8] + a[8]*b[6]) - D2*a[7]*b[7] + D2*a[5]*b[5];
}

// x[a,0,f] = embedding[Z[a], f]; other components zero.  (grid covers A*9*64)
__global__ void init_x_kernel(float* __restrict__ x, const int* __restrict__ zn,
                              const float* __restrict__ emb) {
  const int idx = blockIdx.x * 256 + threadIdx.x;
  const int a = idx / ROWSZ;
  const int r = idx - a * ROWSZ;
  const int d = r >> 6, f = r & 63;
  x[idx] = (d == 0) ? emb[zn[a] * NF + f] : 0.0f;
}

__global__ void zero_kernel(float* __restrict__ p) {
  p[blockIdx.x * 256 + threadIdx.x] = 0.0f;
}

// One wave32 per edge; lane owns features (lane, lane+32). Fused: RBF, filter,
// cutoff, SH, gather x[idx_j] (L2-resident), CG contraction, atomic scatter.
__global__ __launch_bounds__(256) void edge_kernel(
    const float* __restrict__ x, const float* __restrict__ rij,
    const int* __restrict__ idx_i, const int* __restrict__ idx_j,
    const float* __restrict__ fw, const float* __restrict__ fb,
    float* __restrict__ dx) {
  const int lane = threadIdx.x & 31;
  const int e = blockIdx.x * 8 + (threadIdx.x >> 5);
  if (e >= N_EDGES) return;

  const float rx = rij[3*e+0], ry = rij[3*e+1], rz = rij[3*e+2];
  const float d = sqrtf(rx*rx + ry*ry + rz*rz);
  const float inv = 1.0f / d;
  const float X = rx*inv, Yy = ry*inv, Z = rz*inv;

  float Y[9];
  Y[0] = 0.28209479177387814f;
  Y[1] = 0.4886025119029199f * Yy;
  Y[2] = 0.4886025119029199f * Z;
  Y[3] = 0.4886025119029199f * X;
  Y[4] = 1.0925484305920792f * X * Yy;
  Y[5] = 1.0925484305920792f * Yy * Z;
  Y[6] = 0.31539156525252005f * (3.0f * Z * Z - 1.0f);
  Y[7] = 1.0925484305920792f * X * Z;
  Y[8] = 0.5462742152960396f * (X * X - Yy * Yy);

  const float width = CUT_R / (NRBF - 1);
  const float iw = 1.0f / width;
  float rad[NRBF];
#pragma unroll
  for (int k = 0; k < NRBF; ++k) {
    const float u = (d - (float)k * width) * iw;
    rad[k] = expf(-0.5f * u * u);
  }
  const float cut = (d < CUT_R) ? 0.5f * (cosf(3.14159265358979323846f * d / CUT_R) + 1.0f) : 0.0f;

  float Wl[3][2];
#pragma unroll
  for (int l = 0; l < 3; ++l) {
#pragma unroll
    for (int ff = 0; ff < 2; ++ff) {
      const int col = l * 64 + lane + ff * 32;
      float acc = fb[col];
#pragma unroll
      for (int k = 0; k < NRBF; ++k) acc = fmaf(rad[k], fw[k * 192 + col], acc);
      Wl[l][ff] = acc * cut;
    }
  }

  const int j = idx_j[e], i = idx_i[e];
  const float* __restrict__ xj = x + (size_t)j * ROWSZ;
  float* __restrict__ dxi = dx + (size_t)i * ROWSZ;

#pragma unroll
  for (int ff = 0; ff < 2; ++ff) {
    const int f = lane + ff * 32;
    float a[9], b[9], o[9];
#pragma unroll
    for (int q = 0; q < 9; ++q) a[q] = xj[q * 64 + f];
    b[0] = Y[0] * Wl[0][ff];
    b[1] = Y[1] * Wl[1][ff]; b[2] = Y[2] * Wl[1][ff]; b[3] = Y[3] * Wl[1][ff];
    b[4] = Y[4] * Wl[2][ff]; b[5] = Y[5] * Wl[2][ff]; b[6] = Y[6] * Wl[2][ff];
    b[7] = Y[7] * Wl[2][ff]; b[8] = Y[8] * Wl[2][ff];
    cg_contract(a, b, o);
#pragma unroll
    for (int q = 0; q < 9; ++q)
      __hip_atomic_fetch_add(&dxi[q * 64 + f], o[q], __ATOMIC_RELAXED, __HIP_MEMORY_SCOPE_AGENT);
  }
}

// 16 atoms per block, 9 waves (wave c owns spherical component c).
// ddx = dx@m1 (WMMA) -> CG self-mix -> @m2 (WMMA) -> gate GEMM+sigmoid (WMMA)
// -> gated @m3 (WMMA) -> x += result.
__global__ __launch_bounds__(288) void atom_kernel(
    float* __restrict__ x, const float* __restrict__ dx,
    const float* __restrict__ m1w, const float* __restrict__ m2w,
    const float* __restrict__ m3w, const float* __restrict__ gw,
    const float* __restrict__ gb) {
  __shared__ float buf1[16 * ROWSZ];   // 36.9 KB
  __shared__ float buf2[16 * ROWSZ];   // 36.9 KB
  __shared__ float sbuf[16 * 192];     // 12.3 KB  sigmoid gates
  const int tid = threadIdx.x;
  const int wv = tid >> 5;             // 0..8 = spherical component
  const int lane = tid & 31;
  const int hf = lane >> 4;            // half-wave (K split per ISA A-layout)
  const int l15 = lane & 15;
  const int a0 = blockIdx.x * 16;
  const int c = wv;
  const int lidx_c = (c == 0) ? 0 : ((c < 4) ? 1 : 2);

  int gaL = a0 + l15; if (gaL > N_ATOMS - 1) gaL = N_ATOMS - 1;  // clamped loads

  // ---- GEMM1: ddx(slab c) = dx(slab c) @ m1w -> buf1 ----
  {
    v2f afr[16];
    const float* __restrict__ arow = dx + (size_t)gaL * ROWSZ + c * 64;
#pragma unroll
    for (int s = 0; s < 16; ++s) {
      const int k = 4 * s + 2 * hf;
      v2f a; a.x = arow[k]; a.y = arow[k + 1];
      afr[s] = a;
    }
#pragma unroll
    for (int n = 0; n < 4; ++n) {
      v8f acc = {};
#pragma unroll
      for (int s = 0; s < 16; ++s) {
        const int k = 4 * s + 2 * hf;
        v2f b; b.x = m1w[k * 64 + n * 16 + l15]; b.y = m1w[(k + 1) * 64 + n * 16 + l15];
        acc = wmma4(afr[s], b, acc);
      }
#pragma unroll
      for (int v = 0; v < 8; ++v)
        buf1[((v + 8 * hf) * SDIM + c) * 64 + n * 16 + l15] = acc[v];
    }
  }
  __syncthreads();

  // ---- CG self-mix: buf2 = dx + CG(dx, ddx) ----
  for (int idx = tid; idx < 16 * 64; idx += 288) {
    const int al = idx >> 6, f = idx & 63;
    int ga = a0 + al; if (ga > N_ATOMS - 1) ga = N_ATOMS - 1;
    float aa[9], bb[9], oo[9];
#pragma unroll
    for (int q = 0; q < 9; ++q) {
      aa[q] = dx[(size_t)ga * ROWSZ + q * 64 + f];
      bb[q] = buf1[(al * SDIM + q) * 64 + f];
    }
    cg_contract(aa, bb, oo);
#pragma unroll
    for (int q = 0; q < 9; ++q)
      buf2[(al * SDIM + q) * 64 + f] = aa[q] + oo[q];
  }
  __syncthreads();

  // ---- GEMM2: dx3(slab c) = buf2(slab c) @ m2w -> buf1 ----
  {
    v2f afr[16];
#pragma unroll
    for (int s = 0; s < 16; ++s) {
      const int k = 4 * s + 2 * hf;
      v2f a; a.x = buf2[(l15 * SDIM + c) * 64 + k]; a.y = buf2[(l15 * SDIM + c) * 64 + k + 1];
      afr[s] = a;
    }
#pragma unroll
    for (int n = 0; n < 4; ++n) {
      v8f acc = {};
#pragma unroll
      for (int s = 0; s < 16; ++s) {
        const int k = 4 * s + 2 * hf;
        v2f b; b.x = m2w[k * 64 + n * 16 + l15]; b.y = m2w[(k + 1) * 64 + n * 16 + l15];
        acc = wmma4(afr[s], b, acc);
      }
#pragma unroll
      for (int v = 0; v < 8; ++v)
        buf1[((v + 8 * hf) * SDIM + c) * 64 + n * 16 + l15] = acc[v];
    }
  }
  __syncthreads();

  // ---- Gate GEMM: s = sigmoid(dx3[:,0,:] @ gw + gb) -> sbuf (waves 0..5, 2 tiles each) ----
  if (wv < 6) {
    v2f afr[16];
#pragma unroll
    for (int s = 0; s < 16; ++s) {
      const int k = 4 * s + 2 * hf;
      v2f a; a.x = buf1[(l15 * SDIM + 0) * 64 + k]; a.y = buf1[(l15 * SDIM + 0) * 64 + k + 1];
      afr[s] = a;
    }
#pragma unroll
    for (int nn = 0; nn < 2; ++nn) {
      const int n = wv * 2 + nn;
      v8f acc = {};
#pragma unroll
      for (int s = 0; s < 16; ++s) {
        const int k = 4 * s + 2 * hf;
        v2f b; b.x = gw[k * 192 + n * 16 + l15]; b.y = gw[(k + 1) * 192 + n * 16 + l15];
        acc = wmma4(afr[s], b, acc);
      }
#pragma unroll
      for (int v = 0; v < 8; ++v) {
        const int col = n * 16 + l15;
        const float h = acc[v] + gb[col];
        sbuf[(v + 8 * hf) * 192 + col] = 1.0f / (1.0f + expf(-h));
      }
    }
  }
  __syncthreads();

  // ---- GEMM3: x += (dx3 * gate) @ m3w (gating fused into A-fragment load) ----
  {
    v2f afr[16];
#pragma unroll
    for (int s = 0; s < 16; ++s) {
      const int k = 4 * s + 2 * hf;
      const float g0 = sbuf[l15 * 192 + lidx_c * 64 + k];
      const float g1 = sbuf[l15 * 192 + lidx_c * 64 + k + 1];
      v2f a;
      a.x = buf1[(l15 * SDIM + c) * 64 + k] * g0;
      a.y = buf1[(l15 * SDIM + c) * 64 + k + 1] * g1;
      afr[s] = a;
    }
#pragma unroll
    for (int n = 0; n < 4; ++n) {
      v8f acc = {};
#pragma unroll
      for (int s = 0; s < 16; ++s) {
        const int k = 4 * s + 2 * hf;
        v2f b; b.x = m3w[k * 64 + n * 16 + l15]; b.y = m3w[(k + 1) * 64 + n * 16 + l15];
        acc = wmma4(afr[s], b, acc);
      }
#pragma unroll
      for (int v = 0; v < 8; ++v) {
        const int al = v + 8 * hf;
        const int ga = a0 + al;
        if (ga < N_ATOMS) {
          const size_t off = (size_t)ga * ROWSZ + c * 64 + n * 16 + l15;
          x[off] += acc[v];
        }
      }
    }
  }
}

extern "C" void kernel_launch(void* const* d_in, const int* in_sizes, int n_in,
                              void* d_out, int out_size, void* d_ws, size_t ws_size,
                              hipStream_t stream) {
  const int*   zn    = (const int*)d_in[0];
  const float* r_ij  = (const float*)d_in[1];
  const int*   idx_i = (const int*)d_in[2];
  const int*   idx_j = (const int*)d_in[3];
  const float* emb   = (const float*)d_in[4];
  const float* fw    = (const float*)d_in[5];   // (3, 20, 192)
  const float* fb    = (const float*)d_in[6];   // (3, 192)
  const float* m1    = (const float*)d_in[7];   // (3, 64, 64)
  const float* m2    = (const float*)d_in[8];
  const float* m3    = (const float*)d_in[9];
  const float* gw    = (const float*)d_in[10];  // (3, 64, 192)
  const float* gb    = (const float*)d_in[11];  // (3, 192)

  float* x  = (float*)d_out;                    // (A, 9, 64) accumulator
  float* dx = (float*)d_ws;                     // (A, 9, 64) scratch, 11.5 MB

  const int nElem = N_ATOMS * ROWSZ;            // 2,880,000 = 11250 * 256
  init_x_kernel<<<nElem / 256, 256, 0, stream>>>(x, zn, emb);

  for (int t = 0; t < 3; ++t) {
    zero_kernel<<<nElem / 256, 256, 0, stream>>>(dx);
    edge_kernel<<<N_EDGES / 8, 256, 0, stream>>>(
        x, r_ij, idx_i, idx_j, fw + (size_t)t * NRBF * 192, fb + (size_t)t * 192, dx);
    atom_kernel<<<(N_ATOMS + 15) / 16, 288, 0, stream>>>(
        x, dx, m1 + (size_t)t * 64 * 64, m2 + (size_t)t * 64 * 64,
        m3 + (size_t)t * 64 * 64, gw + (size_t)t * 64 * 192, gb + (size_t)t * 192);
  }
}